// SpGraphConvolutionLayer_13374528160094
// MI455X (gfx1250) — compile-verified
//
#include <hip/hip_runtime.h>

#define IN_F 256
#define OUT_F 128
#define PITCH 288   // floats per K-pair row: 288 % 64 == 32 dwords -> conflict-free half-waves

typedef __attribute__((ext_vector_type(2))) float v2f;
typedef __attribute__((ext_vector_type(8))) float v8f;

// ---------------------------------------------------------------------------
// Kernel 0: zero the aggregation buffer (d_out) and the degree buffer.
// d_out/d_ws are poisoned by the harness, so we must re-zero every call.
// ---------------------------------------------------------------------------
__global__ void gcn_zero_kernel(float* __restrict__ out, float* __restrict__ deg,
                                int total_out, int total_deg) {
    int stride = gridDim.x * blockDim.x;
    for (int j = blockIdx.x * blockDim.x + threadIdx.x; j < total_out; j += stride)
        out[j] = 0.0f;
    for (int j = blockIdx.x * blockDim.x + threadIdx.x; j < total_deg; j += stride)
        deg[j] = 0.0f;
}

// ---------------------------------------------------------------------------
// Kernel 1: h = X @ W  via V_WMMA_F32_16X16X4_F32 (exact fp32 accumulate).
// Block = 256 threads = 8 waves. Each wave owns a 16-row strip and computes
// all 128 output columns (8 N-tiles of 16) so X is streamed exactly once.
// W staged in LDS K-pair interleaved: wlds[p][n] = {W[2p][n], W[2p+1][n]},
// so each B fragment is ONE aligned ds_load_b64 straight into a VGPR pair.
// ---------------------------------------------------------------------------
__global__ __launch_bounds__(256) void gcn_gemm_wmma(const float* __restrict__ X,
                                                     const float* __restrict__ W,
                                                     float* __restrict__ H,
                                                     int n) {
    __shared__ float wlds[(IN_F / 2) * PITCH];   // 144 KB (< 320 KB per WGP)

    const int tid = threadIdx.x;

    // Cooperative interleaved W stage: 128 pair-rows x 32 col-quads.
    // Two global float4 loads (rows 2p, 2p+1) -> two interleaved b128 LDS stores.
    {
        const float4* __restrict__ wsrc = (const float4*)W;   // [256][32] float4
        for (int i = tid; i < (IN_F / 2) * 32; i += 256) {
            int p = i >> 5;          // K-pair row
            int q = i & 31;          // column quad (4 cols)
            float4 r0 = wsrc[(2 * p + 0) * 32 + q];
            float4 r1 = wsrc[(2 * p + 1) * 32 + q];
            float* dst = &wlds[p * PITCH + q * 8];
            *(float4*)(dst + 0) = make_float4(r0.x, r1.x, r0.y, r1.y);
            *(float4*)(dst + 4) = make_float4(r0.z, r1.z, r0.w, r1.w);
        }
    }
    __syncthreads();

    const int wave = tid >> 5;
    const int lane = tid & 31;
    const int m    = lane & 15;          // row-in-tile for A, col-in-tile for B/C/D
    const int kp   = (lane >> 4) << 1;   // K offset: 0 (lanes 0-15) or 2 (lanes 16-31)

    const int row0 = (blockIdx.x * 8 + wave) * 16;

    // Clamp A row for a ragged last tile; garbage rows are simply not stored.
    int arow = row0 + m;
    if (arow >= n) arow = n - 1;
    const float* __restrict__ xrow = X + (long)arow * IN_F;

    v8f acc[8] = {};   // 8 N-tiles x 16x16 f32 accumulators (64 VGPRs)

    for (int kk = 0; kk < IN_F; kk += 4) {
        // A fragment: 16x4 tile, per-lane float2 (K = kp, kp+1) -> global_load_b64
        v2f a;
        a.x = xrow[kk + kp + 0];
        a.y = xrow[kk + kp + 1];

        // Pair row for this half-wave: (kk + kp) / 2
        const float* __restrict__ wp = &wlds[((kk + kp) >> 1) * PITCH];

#pragma unroll
        for (int t = 0; t < 8; ++t) {
            // B fragment: single ds_load_b64, {W[k][n], W[k+1][n]} pre-paired
            v2f b = *(const v2f*)&wp[(t * 16 + m) * 2];
            acc[t] = __builtin_amdgcn_wmma_f32_16x16x4_f32(
                /*neg_a=*/false, a, /*neg_b=*/false, b,
                /*c_mod=*/(short)0, acc[t],
                /*reuse_a=*/false, /*reuse_b=*/false);
        }
    }

    // C/D layout: lanes 0-15 -> M = v, lanes 16-31 -> M = v + 8; N = lane&15.
    const int rhalf = (lane >> 4) * 8;
    if (row0 + 16 <= n) {
        // Wave-uniform fast path (always taken when n % 16 == 0): one clause of
        // 64 global_store_b32 with immediate offsets off a single base pointer.
        float* __restrict__ hp = H + (long)(row0 + rhalf) * OUT_F + m;
#pragma unroll
        for (int t = 0; t < 8; ++t) {
#pragma unroll
            for (int v = 0; v < 8; ++v)
                hp[v * OUT_F + t * 16] = acc[t][v];
        }
    } else {
#pragma unroll
        for (int t = 0; t < 8; ++t) {
#pragma unroll
            for (int v = 0; v < 8; ++v) {
                int r = row0 + rhalf + v;
                if (r < n)
                    H[(long)r * OUT_F + t * 16 + m] = acc[t][v];
            }
        }
    }
}

// ---------------------------------------------------------------------------
// Kernel 2: edge scatter. One wave32 per edge: each lane gathers a float4 of
// h[col[e]] (global_load_b128) and issues 4 hardware f32 atomics into
// out[row[e]]. Lane 0 accumulates the degree. h and out both fit in the
// 192 MB L2, so this is L2-atomic-throughput bound — the intended path.
// ---------------------------------------------------------------------------
__global__ __launch_bounds__(256) void gcn_scatter_kernel(const float* __restrict__ H,
                                                          const int* __restrict__ row,
                                                          const int* __restrict__ col,
                                                          float* __restrict__ out,
                                                          float* __restrict__ deg,
                                                          int e) {
    const int lane  = threadIdx.x & 31;
    const int wid   = (blockIdx.x * blockDim.x + threadIdx.x) >> 5;
    const int nwave = (gridDim.x * blockDim.x) >> 5;

    for (int ed = wid; ed < e; ed += nwave) {
        const int r = row[ed];
        const int c = col[ed];

        const float4 hv = ((const float4*)(H + (long)c * OUT_F))[lane];
        float* o = out + (long)r * OUT_F + lane * 4;

        unsafeAtomicAdd(o + 0, hv.x);
        unsafeAtomicAdd(o + 1, hv.y);
        unsafeAtomicAdd(o + 2, hv.z);
        unsafeAtomicAdd(o + 3, hv.w);

        if (lane == 0)
            unsafeAtomicAdd(deg + r, 1.0f);
    }
}

// ---------------------------------------------------------------------------
// Kernel 3: out[r][f] /= max(deg[r], 1)   (zero-degree rows stay 0).
// ---------------------------------------------------------------------------
__global__ void gcn_finalize_kernel(float* __restrict__ out,
                                    const float* __restrict__ deg,
                                    int total, int n) {
    int i = blockIdx.x * blockDim.x + threadIdx.x;
    if (i >= total) return;
    float d = deg[i >> 7];               // OUT_F == 128
    out[i] = out[i] / (d > 0.0f ? d : 1.0f);
}

// ---------------------------------------------------------------------------
extern "C" void kernel_launch(void* const* d_in, const int* in_sizes, int n_in,
                              void* d_out, int out_size, void* d_ws, size_t ws_size,
                              hipStream_t stream) {
    const float* x    = (const float*)d_in[0];   // [n, 256] fp32
    const int*   edge = (const int*)d_in[1];     // [2, e] int32 (harness int)
    const float* W    = (const float*)d_in[2];   // [256, 128] fp32
    float*       out  = (float*)d_out;           // [n, 128] fp32

    const int n = in_sizes[0] / IN_F;            // 100000
    const int e = in_sizes[1] / 2;               // 1600000
    const int* row = edge;                       // edge[0, :]
    const int* col = edge + e;                   // edge[1, :]

    // Workspace layout: h [n*128 f32] | deg [n f32]
    float* h   = (float*)d_ws;
    float* deg = h + (size_t)n * OUT_F;

    const int total = n * OUT_F;

    gcn_zero_kernel<<<1024, 256, 0, stream>>>(out, deg, total, n);

    const int gemm_blocks = (n + 127) / 128;     // 128 rows per block (8 waves x 16)
    gcn_gemm_wmma<<<gemm_blocks, 256, 0, stream>>>(x, W, h, n);

    gcn_scatter_kernel<<<4096, 256, 0, stream>>>(h, row, col, out, deg, e);

    gcn_finalize_kernel<<<(total + 255) / 256, 256, 0, stream>>>(out, deg, total, n);
}